// DiagonalRefine_42554535969020
// MI455X (gfx1250) — compile-verified
//
#include <hip/hip_runtime.h>

typedef __attribute__((ext_vector_type(2))) float v2f;
typedef __attribute__((ext_vector_type(4))) float v4f;
typedef __attribute__((ext_vector_type(8))) float v8f;

#define BB 2
#define CC 256
#define HH 512
#define WW 512
#define LL 512
#define KK (CC * 3)   // 768

__device__ __forceinline__ float gelu_exact(float x) {
    // exact GELU: 0.5*x*(1+erf(x/sqrt(2)))
    return 0.5f * x * (1.0f + erff(x * 0.70710678118654752440f));
}

// -------- 1. gather the diagonal: diag[b,c,l] = feat[b,c,l,l] --------
__global__ void diag_gather(const float* __restrict__ feat, float* __restrict__ diag) {
    int idx = blockIdx.x * blockDim.x + threadIdx.x;     // BB*CC*LL threads
    int l  = idx & (LL - 1);
    int bc = idx >> 9;
    diag[idx] = feat[((size_t)bc * HH + l) * WW + l];
}

// -------- 2. grouped conv1d (8 groups of 32) + bias + exact GELU --------
__global__ void gconv1(const float* __restrict__ diag, const float* __restrict__ w1,
                       const float* __restrict__ b1, float* __restrict__ x1) {
    int idx = blockIdx.x * blockDim.x + threadIdx.x;     // BB*CC*LL threads
    int l  = idx & (LL - 1);
    int co = (idx >> 9) & (CC - 1);
    int b  = idx >> 17;
    int g  = co >> 5;                                    // group = co/32
    const float* wp = w1 + co * 32 * 3;
    const float* dp = diag + (b * CC + g * 32) * LL;
    float s = b1[co];
    #pragma unroll
    for (int ci = 0; ci < 32; ++ci) {
        const float* d = dp + ci * LL;
        float xm = (l > 0)      ? d[l - 1] : 0.0f;
        float x0 = d[l];
        float xp = (l < LL - 1) ? d[l + 1] : 0.0f;
        s = fmaf(wp[ci * 3 + 0], xm, s);
        s = fmaf(wp[ci * 3 + 1], x0, s);
        s = fmaf(wp[ci * 3 + 2], xp, s);
    }
    x1[idx] = gelu_exact(s);
}

// -------- 3a. im2col: Bmat[b, ci*3+k, l] = x1pad[b, ci, l+k-1] --------
// Moves all padding logic out of the WMMA hot loop.
__global__ void im2col(const float* __restrict__ x1, float* __restrict__ Bmat) {
    int idx = blockIdx.x * blockDim.x + threadIdx.x;     // BB*KK*LL threads
    int l    = idx & (LL - 1);
    int t    = idx >> 9;
    int kidx = t & (KK - 1);                             // KK=768 is not pow2-safe...
    // KK = 768 = 0x300, not a power of two: use explicit mod/div instead.
    kidx = t % KK;
    int b    = t / KK;
    int ci = kidx / 3;
    int k  = kidx - ci * 3;
    int lsrc = l + k - 1;
    float v = (lsrc >= 0 && lsrc < LL) ? x1[(b * CC + ci) * LL + lsrc] : 0.0f;
    Bmat[idx] = v;
}

// -------- 3b. dense conv1d as GEMM via V_WMMA_F32_16X16X4_F32 --------
// Per batch: OUT[256 x 512] = A[256 x 768] * Bmat[768 x 512]
//   A[co, kidx] = w2[co*768 + kidx]  (contiguous rows)
// 4 waves per block, one 16x16 output tile per wave, K stepped by 4.
__global__ void dense_wmma(const float* __restrict__ Bmat, const float* __restrict__ w2,
                           const float* __restrict__ b2, float* __restrict__ x2) {
    int wave = (blockIdx.x << 2) + (threadIdx.x >> 5);   // global tile id
    int nt = wave & 31;                                  // L/16 = 32 tiles
    int mt = (wave >> 5) & 15;                           // C/16 = 16 tiles
    int b  = wave >> 9;
    int lane = threadIdx.x & 31;
    int half = lane >> 4;                                // 0: K pair (0,1); 1: K pair (2,3)
    int lid  = lane & 15;
    int m0 = mt << 4, n0 = nt << 4;

    const float* Arow = w2 + (size_t)(m0 + lid) * KK;    // weight row, contiguous in kidx
    const float* Bcol = Bmat + (size_t)b * KK * LL + (n0 + lid);

    v8f acc = {};
    #pragma unroll 4
    for (int kb = 0; kb < KK; kb += 4) {
        int k0 = kb + (half << 1);
        // A fp32 16x4: lanes 0-15 -> (K=kb,kb+1); lanes 16-31 -> (K=kb+2,kb+3)
        v2f a = *(const v2f*)(Arow + k0);                // 8B-aligned b64 load
        // B fp32 4x16 (mirrored): lane supplies B[k0, n], B[k0+1, n]
        v2f bv;
        bv.x = Bcol[(size_t)k0 * LL];
        bv.y = Bcol[(size_t)(k0 + 1) * LL];
        acc = __builtin_amdgcn_wmma_f32_16x16x4_f32(
            /*neg_a=*/false, a, /*neg_b=*/false, bv,
            /*c_mod=*/(short)0, acc, /*reuse_a=*/false, /*reuse_b=*/false);
    }

    // C/D layout: VGPR r -> M = r + 8*half, N = lid
    #pragma unroll
    for (int r = 0; r < 8; ++r) {
        int co = m0 + r + (half << 3);
        int l  = n0 + lid;
        float v = gelu_exact(acc[r] + b2[co]);
        x2[((size_t)b * CC + co) * LL + l] = v;
    }
}

// -------- 4. fused diag_embed + depthwise 3x3 blur --------
// out[b,c,i,j] = sum_ki w_blur[c,0,ki,ki+(i-j)] * x2[b,c,i+ki-1]  for |i-j|<=2, else 0.
// One float4 non-temporal store per thread covers the whole 512MB output.
__global__ void fused_out(const float* __restrict__ x2, const float* __restrict__ w_blur,
                          float* __restrict__ out) {
    int idx = blockIdx.x * blockDim.x + threadIdx.x;     // BB*CC*HH*(WW/4) threads
    int jb = idx & 127;                                  // W/4 = 128
    int t  = idx >> 7;
    int i  = t & (HH - 1);
    int bc = t >> 9;
    int c  = bc & (CC - 1);
    int j0 = jb << 2;

    v4f v = {0.0f, 0.0f, 0.0f, 0.0f};
    // float4 block [j0, j0+3] intersects band [i-2, i+2]?
    if (j0 + 5 >= i && j0 <= i + 2) {
        const float* wb = w_blur + c * 9;
        const float* xr = x2 + (size_t)bc * LL;
        #pragma unroll
        for (int jj = 0; jj < 4; ++jj) {
            int j = j0 + jj;
            int d = i - j;
            if (d >= -2 && d <= 2) {
                float s = 0.0f;
                #pragma unroll
                for (int ki = 0; ki < 3; ++ki) {
                    int kj = ki + d;
                    int r  = i + ki - 1;
                    if (kj >= 0 && kj <= 2 && r >= 0 && r < HH)
                        s = fmaf(wb[ki * 3 + kj], xr[r], s);
                }
                v[jj] = s;
            }
        }
    }
    __builtin_nontemporal_store(v, (v4f*)out + idx);
}

extern "C" void kernel_launch(void* const* d_in, const int* in_sizes, int n_in,
                              void* d_out, int out_size, void* d_ws, size_t ws_size,
                              hipStream_t stream) {
    const float* feat   = (const float*)d_in[0];
    const float* w1     = (const float*)d_in[1];
    const float* b1     = (const float*)d_in[2];
    const float* w2     = (const float*)d_in[3];
    const float* b2     = (const float*)d_in[4];
    const float* w_blur = (const float*)d_in[5];
    float* out = (float*)d_out;

    float* diag = (float*)d_ws;                   // 262144 floats (1 MB)
    float* x1   = diag + BB * CC * LL;            // 262144 floats (1 MB)
    float* x2   = x1   + BB * CC * LL;            // 262144 floats (1 MB)
    float* Bmat = x2   + BB * CC * LL;            // BB*768*512 = 786432 floats (3 MB)

    const int n1 = BB * CC * LL;                  // 262144
    diag_gather<<<n1 / 256, 256, 0, stream>>>(feat, diag);
    gconv1<<<n1 / 256, 256, 0, stream>>>(diag, w1, b1, x1);

    const int n2 = BB * KK * LL;                  // 786432
    im2col<<<n2 / 256, 256, 0, stream>>>(x1, Bmat);

    // 1024 tiles total, 4 waves (tiles) per 128-thread block
    dense_wmma<<<(BB * (CC / 16) * (LL / 16)) / 4, 128, 0, stream>>>(Bmat, w2, b2, x2);

    const int n4 = BB * CC * HH * (WW / 4);       // 33554432 float4 stores
    fused_out<<<n4 / 256, 256, 0, stream>>>(x2, w_blur, out);
}